// VectorQuantizer_11106785427820
// MI455X (gfx1250) — compile-verified
//
#include <hip/hip_runtime.h>
#include <hip/hip_bf16.h>

// ---------------------------------------------------------------------------
// VQ-VAE vector quantizer for MI455X (gfx1250, wave32, WMMA + async-to-LDS).
//   z_e      : [N=65536, D=64] f32
//   codebook : [K=8192,  D=64] f32
// Outputs (concatenated in d_out, f32):
//   z_q_st [N*D] | vq_loss [1] | encoding_indices [N] (stored as float)
//
// Score GEMM uses V_WMMA_F32_16X16X4_F32 with two independent accumulator
// chains; -2 scale is folded (exactly, power of two) into the resident B
// operand (z_e rows), so codebook tiles are raw-copied global->LDS with
// GLOBAL_LOAD_ASYNC_TO_LDS_B128 (per-lane scatter into a bank-conflict-free
// 68-float padded layout), double-buffered and synced via s_wait_asynccnt.
// e2[k] is folded into the GEMM as an extra K-slice (A-ext=e2, B-ext=1).
// ---------------------------------------------------------------------------

typedef float  v2f __attribute__((ext_vector_type(2)));
typedef float  v8f __attribute__((ext_vector_type(8)));

#define N_ROWS      65536
#define K_CB        8192
#define D_DIM       64
#define TILE_K      16
#define NUM_TILES   (K_CB / TILE_K)      // 512
#define WAVES_PB    8
#define ROWS_PB     (WAVES_PB * 16)      // 128 z_e rows per block
#define NUM_BLOCKS  (N_ROWS / ROWS_PB)   // 512
#define LDS_STRIDE  68                   // padded row (bank-conflict-free, 16B-aligned)
#define NUM_PARTIALS (NUM_BLOCKS * WAVES_PB)  // 4096

// ---- CDNA5 async global->LDS copy (ASYNCcnt-tracked), inline asm ----------
__device__ __forceinline__ void async_copy_b128_to_lds(unsigned lds_addr,
                                                       const float* gptr) {
    asm volatile("global_load_async_to_lds_b128 %0, %1, off"
                 :: "v"(lds_addr), "v"((unsigned long long)(uintptr_t)gptr)
                 : "memory");
}
__device__ __forceinline__ void wait_asynccnt0() {
    asm volatile("s_wait_asynccnt 0" ::: "memory");
}
__device__ __forceinline__ unsigned lds_addr_of(const float* p) {
    return (unsigned)(uintptr_t)p;   // flat-LDS: addr[31:0] is the LDS address
}

// ---------------- prep: e2[k] = sum_d codebook[k][d]^2 ----------------------
__global__ void vq_e2_kernel(const float* __restrict__ cb, float* __restrict__ e2) {
    int wave = threadIdx.x >> 5;
    int lane = threadIdx.x & 31;
    int row  = blockIdx.x * 8 + wave;
    const float* r = cb + (size_t)row * D_DIM;
    float a = r[lane], b = r[lane + 32];
    float s = a * a + b * b;
    #pragma unroll
    for (int off = 16; off; off >>= 1) s += __shfl_xor(s, off, 32);
    if (lane == 0) e2[row] = s;
}

// ---------------- main: fused score GEMM + argmin + gather + loss -----------
__global__ void __launch_bounds__(256)
vq_main_kernel(const float* __restrict__ ze, const float* __restrict__ cb,
               const float* __restrict__ e2, float* __restrict__ zq_out,
               float* __restrict__ idx_out, float* __restrict__ partials) {
    __shared__ float cbs[2][TILE_K * LDS_STRIDE];
    __shared__ float e2s[2][TILE_K];

    const int tid   = threadIdx.x;
    const int wave  = tid >> 5;
    const int lane  = tid & 31;
    const int lhalf = lane >> 4;       // 0: lanes 0-15, 1: lanes 16-31
    const int lrow  = lane & 15;       // N index (z_e row) and A's M index
    const int rowbase = blockIdx.x * ROWS_PB + wave * 16;

    // ---- B operands: 16 z_e rows (pre-scaled by -2, exact) in registers ----
    // B layout (4x16, K x N): lane n=lrow; VGPR0 holds K={0 | 2}, VGPR1 K={1 | 3}
    v2f b[16];
    const float* zrow = ze + ((size_t)(rowbase + lrow)) * D_DIM + lhalf * 2;
    #pragma unroll
    for (int kk = 0; kk < 16; ++kk) {
        v2f t = *(const v2f*)(zrow + kk * 4);
        b[kk].x = -2.0f * t.x;
        b[kk].y = -2.0f * t.y;
    }
    v2f bext;                           // appended K-slice: B[64][n]=1, rest 0
    bext.x = (lhalf == 0) ? 1.0f : 0.0f;
    bext.y = 0.0f;

    float bestv = 3.0e38f;
    int   besti = 0;

    // ---- staging geometry: 256 lanes x 16B cover one 16x64 f32 tile -------
    const int srow = tid >> 4;          // 0..15 codebook row within tile
    const int scol = (tid & 15) * 4;    // float column (groups of 4)
    const unsigned ldsA0 = lds_addr_of(&cbs[0][srow * LDS_STRIDE + scol]);
    const unsigned ldsA1 = lds_addr_of(&cbs[1][srow * LDS_STRIDE + scol]);

    // ---- preamble: async-copy tile 0 into buffer 0 -------------------------
    async_copy_b128_to_lds(ldsA0, cb + (size_t)srow * D_DIM + scol);
    {
        float v0 = (tid < TILE_K) ? e2[tid] : 0.0f;
        wait_asynccnt0();
        if (tid < TILE_K) e2s[0][tid] = v0;
    }
    __syncthreads();

    for (int t = 0; t < NUM_TILES; ++t) {
        const int  cur      = t & 1;
        const bool has_next = (t + 1 < NUM_TILES);

        // issue async copy of next tile into the other buffer (overlaps WMMA)
        float ne2 = 0.0f;
        if (has_next) {
            const float* gsrc = cb + ((size_t)(t + 1) * TILE_K + srow) * D_DIM + scol;
            async_copy_b128_to_lds(cur ? ldsA0 : ldsA1, gsrc);
            if (tid < TILE_K) ne2 = e2[(t + 1) * TILE_K + tid];
            if (t + 8 < NUM_TILES)
                __builtin_prefetch(cb + ((size_t)(t + 8) * TILE_K + srow) * D_DIM + scol, 0, 1);
        }

        // ---- 17 x V_WMMA_F32_16X16X4_F32, two independent chains ----------
        v8f c0 = {}, c1 = {};
        const float* arow = &cbs[cur][lrow * LDS_STRIDE + lhalf * 2];
        #pragma unroll
        for (int kk = 0; kk < 8; ++kk) {
            v2f a0 = *(const v2f*)(arow + kk * 4);
            v2f a1 = *(const v2f*)(arow + (kk + 8) * 4);
            c0 = __builtin_amdgcn_wmma_f32_16x16x4_f32(
                     false, a0, false, b[kk],     (short)0, c0, false, false);
            c1 = __builtin_amdgcn_wmma_f32_16x16x4_f32(
                     false, a1, false, b[kk + 8], (short)0, c1, false, false);
        }
        {   // e2 fold-in on chain 0: A[m][64]=e2[m], B[64][n]=1
            float e2v = e2s[cur][lrow];
            v2f a; a.x = (lhalf == 0) ? e2v : 0.0f; a.y = 0.0f;
            c0 = __builtin_amdgcn_wmma_f32_16x16x4_f32(
                     false, a, false, bext, (short)0, c0, false, false);
        }

        // ---- running argmin (first-min semantics: strict <) ---------------
        #pragma unroll
        for (int j = 0; j < 8; ++j) {
            float s = c0[j] + c1[j];
            int   m = t * TILE_K + lhalf * 8 + j;   // codebook index
            bool better = s < bestv;
            bestv = better ? s : bestv;
            besti = better ? m : besti;
        }

        // ---- close out staging of next tile -------------------------------
        if (has_next) {
            wait_asynccnt0();                       // our wave's async slice done
            if (tid < TILE_K) e2s[cur ^ 1][tid] = ne2;
        }
        __syncthreads();                            // all slices visible block-wide
    }

    // ---- merge the two M-halves (lane l <-> l^16 hold same z_e row) -------
    {
        float ov = __shfl_xor(bestv, 16, 32);
        int   oi = __shfl_xor(besti, 16, 32);
        bool take = (ov < bestv) || (ov == bestv && oi < besti);
        bestv = take ? ov : bestv;
        besti = take ? oi : besti;
    }

    if (lhalf == 0)
        idx_out[rowbase + lrow] = (float)besti;

    // ---- gather z_q rows + deterministic loss partial ----------------------
    float lsum = 0.0f;
    for (int r = 0; r < 16; ++r) {
        int bi  = __shfl(besti, r, 32);
        size_t row = (size_t)(rowbase + r);
        const v2f q = *(const v2f*)(cb + (size_t)bi * D_DIM + 2 * lane);
        const v2f z = *(const v2f*)(ze + row * D_DIM + 2 * lane);
        *(v2f*)(zq_out + row * D_DIM + 2 * lane) = q;
        float d0 = q.x - z.x, d1 = q.y - z.y;
        lsum += d0 * d0 + d1 * d1;
    }
    #pragma unroll
    for (int off = 16; off; off >>= 1) lsum += __shfl_xor(lsum, off, 32);
    if (lane == 0) partials[blockIdx.x * WAVES_PB + wave] = lsum;
}

// ---------------- final: deterministic fixed-order loss reduction -----------
__global__ void vq_loss_kernel(const float* __restrict__ partials,
                               float* __restrict__ out_loss) {
    __shared__ float red[256];
    float s = 0.0f;
    for (int i = threadIdx.x; i < NUM_PARTIALS; i += 256) s += partials[i];
    red[threadIdx.x] = s;
    __syncthreads();
    for (int off = 128; off; off >>= 1) {
        if ((int)threadIdx.x < off) red[threadIdx.x] += red[threadIdx.x + off];
        __syncthreads();
    }
    if (threadIdx.x == 0)
        out_loss[0] = 1.25f * red[0] / (float)((size_t)N_ROWS * D_DIM);
}

// ---------------------------------------------------------------------------
extern "C" void kernel_launch(void* const* d_in, const int* in_sizes, int n_in,
                              void* d_out, int out_size, void* d_ws, size_t ws_size,
                              hipStream_t stream) {
    const float* ze = (const float*)d_in[0];     // [N, 64]
    const float* cb = (const float*)d_in[1];     // [K, 64]

    float* out      = (float*)d_out;
    float* zq_out   = out;                                   // [N*64]
    float* loss_out = out + (size_t)N_ROWS * D_DIM;          // [1]
    float* idx_out  = loss_out + 1;                          // [N]

    float* e2       = (float*)d_ws;                          // [8192]
    float* partials = e2 + K_CB;                             // [4096]

    vq_e2_kernel<<<K_CB / 8, 256, 0, stream>>>(cb, e2);
    vq_main_kernel<<<NUM_BLOCKS, 256, 0, stream>>>(ze, cb, e2, zq_out, idx_out, partials);
    vq_loss_kernel<<<1, 256, 0, stream>>>(partials, loss_out);
}